// EuclidLoss_64862596104222
// MI455X (gfx1250) — compile-verified
//
#include <hip/hip_runtime.h>
#include <math.h>

// CDNA5 / gfx1250, wave32. theta-einsum mapped onto V_WMMA_F32_16X16X4_F32 (K=4 exact fit).
// Branchless early-exit bookkeeping to keep EXEC full and avoid bool-packing VALU noise.

typedef __attribute__((ext_vector_type(2))) float v2f;
typedef __attribute__((ext_vector_type(8))) float v8f;

#define IMG_SIZE 512
#define NCHUNK   32      // 512 r-steps / 16 per WMMA tile
#define THRESH   160.0f

__global__ __launch_bounds__(256) void
euclid_wmma_kernel(const float* __restrict__ coef,   // [B,4]
                   const float* __restrict__ image,  // [512,512] row-major
                   float* __restrict__ out,          // [B]
                   int nB)
{
    const int lane    = threadIdx.x & 31;
    const int wave    = threadIdx.x >> 5;
    const int bbase   = (blockIdx.x * 8 + wave) * 16;   // 16 batches per wave
    if (bbase >= nB) return;                            // uniform per wave

    const bool half_up = (lane >= 16);
    const int  myidx   = lane & 15;

    // --- A tile (16x4 f32, 2 VGPRs): lanes 0-15 -> K=0,1 ; lanes 16-31 -> K=2,3
    const float* arow = coef + (size_t)(bbase + myidx) * 4 + (half_up ? 2 : 0);
    v2f A;
    A.x = arow[0];
    A.y = arow[1];

    float runmin[8];
#pragma unroll
    for (int v = 0; v < 8; ++v) runmin[v] = INFINITY;
    unsigned doneMask = 0;               // bit v: batch row (for my half) terminated

    const float rf0 = (float)myidx;

    for (int chunk = 0; chunk < NCHUNK; ++chunk) {
        const float rf = rf0 + (float)(chunk << 4);      // this lane's r (column N of D)

        // --- B tile (4x16 f32, 2 VGPRs): lanes 0-15 -> rows K0=1, K1=r ; lanes 16-31 -> K2=r^2, K3=r^3
        v2f Bm;
        if (!half_up) { Bm.x = 1.0f;  Bm.y = rf; }
        else          { float r2 = rf * rf; Bm.x = r2; Bm.y = r2 * rf; }

        v8f cz = {};
        // D[batch][r] = coef[16x4] x powers^T[4x16]  (one instruction = 256 thetas)
        v8f theta = __builtin_amdgcn_wmma_f32_16x16x4_f32(
            /*neg_a=*/false, A, /*neg_b=*/false, Bm,
            /*c_mod=*/(short)0, cz, /*reuse_a=*/false, /*reuse_b=*/false);

#pragma unroll
        for (int v = 0; v < 8; ++v) {
            float th = theta[v];                 // batch = bbase + v (+8 if upper half), r = rf
            float s, c;
            __sincosf(th, &s, &c);
            float x  = 256.0f + rf * c;
            float y  = 256.0f + rf * s;
            float dx = x - 400.0f;
            float dy = y - 300.0f;
            float dist = __fsqrt_rn(dx * dx + dy * dy);

            int xi = (int)x; xi = min(max(xi, 0), IMG_SIZE - 1);   // trunc-toward-zero then clip
            int yi = (int)y; yi = min(max(yi, 0), IMG_SIZE - 1);
            bool hit = image[xi * IMG_SIZE + yi] < THRESH;

            // wave32 ballot; pick this lane's 16-wide half (two batch rows per D VGPR)
            unsigned hm    = (unsigned)__ballot(hit);
            unsigned halfm = half_up ? (hm >> 16) : (hm & 0xFFFFu);

            // Sentinel ctz: halfm==0 -> first=16 -> every lane (myidx<=15) contributes dist.
            // halfm!=0 -> only r <= exit_idx contribute (matches run_min at first hit).
            int   first    = __builtin_ctz(halfm | 0x10000u);
            bool  active   = ((doneMask >> v) & 1u) == 0u;
            float contrib  = (myidx <= first) ? dist : INFINITY;
            contrib        = active ? contrib : INFINITY;
            runmin[v]      = fminf(runmin[v], contrib);
            doneMask      |= (halfm ? 1u : 0u) << v;
        }

        // expected to fire after chunk 0 (P ~ 1 - 1.5e-7): ~32x less work than the reference
        if (__all(doneMask == 0xFFu)) break;
    }

    // --- per-batch min across the 16 lanes of each half (xor masks stay inside the half)
#pragma unroll
    for (int v = 0; v < 8; ++v) {
        float m = runmin[v];
        m = fminf(m, __shfl_xor(m, 8));
        m = fminf(m, __shfl_xor(m, 4));
        m = fminf(m, __shfl_xor(m, 2));
        m = fminf(m, __shfl_xor(m, 1));
        if (myidx == 0)
            out[bbase + v + (half_up ? 8 : 0)] = m;
    }
}

extern "C" void kernel_launch(void* const* d_in, const int* in_sizes, int n_in,
                              void* d_out, int out_size, void* d_ws, size_t ws_size,
                              hipStream_t stream)
{
    const float* coef  = (const float*)d_in[0];   // [B,4] f32
    const float* image = (const float*)d_in[1];   // [512,512] f32
    float*       out   = (float*)d_out;           // [B] f32

    const int nB     = in_sizes[0] / 4;           // 32768
    const int blocks = (nB + 127) / 128;          // 128 batches per 256-thread block (8 waves x 16)
    euclid_wmma_kernel<<<blocks, 256, 0, stream>>>(coef, image, out, nB);
}